// ViT_40836549050936
// MI455X (gfx1250) — compile-verified
//
#include <hip/hip_runtime.h>

typedef float v2f __attribute__((ext_vector_type(2)));
typedef float v8f __attribute__((ext_vector_type(8)));
typedef __attribute__((address_space(3))) float lds_float;

#define MT       64
#define KC       32
#define LDS_STR  36                   // LDS tile row stride (floats): 16B-aligned rows, conflict-free b64 frag reads
#define LN_EPS   1e-5f
#define LOG1E4   9.210340371976184f   // ln(10000)

// LDS byte offset (relative to workgroup LDS base) of a __shared__ element.
__device__ __forceinline__ unsigned lds_off(float* p) {
    return (unsigned)(uintptr_t)(lds_float*)p;
}

// CDNA5 async copy: global memory -> LDS, 16 bytes per lane, tracked by ASYNCcnt.
__device__ __forceinline__ void async_load_b128(unsigned lds_byte_off, const float* gaddr) {
    asm volatile("global_load_async_to_lds_b128 %0, %1, off"
                 :: "v"(lds_byte_off), "v"(gaddr)
                 : "memory");
}

__device__ __forceinline__ void wait_async0() {
    asm volatile("s_wait_asynccnt 0x0" ::: "memory");
}

// ---------------------------------------------------------------------------
// Fused: patchify -> GEMM (fp32 WMMA 16x16x4) -> +bias +pos-emb -> LayerNorm
// Grid: 25088/64 = 392 blocks, 512 threads (16 waves, wave32).
// Wave grid 2(M) x 8(N); each wave owns a 32x96 output block (12 WMMA tiles).
// Tiles staged with GLOBAL_LOAD_ASYNC_TO_LDS_B128 (ASYNCcnt path).
// ---------------------------------------------------------------------------
__global__ __launch_bounds__(512)
void vit_patch_gemm_ln(const float* __restrict__ img,
                       const float* __restrict__ Wm,
                       const float* __restrict__ bias,
                       const float* __restrict__ gamma,
                       const float* __restrict__ beta,
                       float* __restrict__ out)
{
    __shared__ float Al[MT * LDS_STR];      //   9,216 B : A tile (64 x 32)
    __shared__ float Bl[768 * LDS_STR];     // 110,592 B : W tile (768 rows x 32 k)
    __shared__ float rowS[MT * 8];          // per-(row, n-wave) partial sums
    __shared__ float rowQ[MT * 8];          // per-(row, n-wave) partial sum-sq
    __shared__ float rowMean[MT];
    __shared__ float rowInv[MT];

    const int tid    = threadIdx.x;
    const int lane   = tid & 31;
    const int wave   = tid >> 5;
    const int lanelo = lane & 15;
    const int laneK  = (lane >> 4) << 1;    // 0 or 2 : K offset of this half-wave
    const int laneM8 = (lane >> 4) << 3;    // 0 or 8 : M offset in C/D layout

    const int wm    = wave & 1;             // M sub-block (rows wm*32 .. +32)
    const int wn    = wave >> 1;            // N sub-block (cols wn*96 .. +96)
    const int mrow0 = wm * 32;
    const int ncol0 = wn * 96;

    const int m0g = blockIdx.x * MT;        // global patch-row base

    // per-column constants for this wave's 6 N-tiles
    float g_r[6], be_r[6], bi_r[6], fr_r[6];
#pragma unroll
    for (int nt = 0; nt < 6; ++nt) {
        int n = ncol0 + nt * 16 + lanelo;
        g_r[nt]  = gamma[n];
        be_r[nt] = beta[n];
        bi_r[nt] = bias[n];
        fr_r[nt] = __expf(-(float)((n >> 1) << 1) * (LOG1E4 / 768.0f)); // 10000^(-2*floor(n/2)/768)
    }

    // A-tile source: this thread stages one 16B packet per k-chunk.
    // patch feature k = c*256 + dy*16 + dx  (C,ph,pw flatten order)
    const int am = tid >> 3;                // local patch row 0..63
    const int aq = tid & 7;                 // which float4 within 32 floats
    const int gp = m0g + am;                // global patch index
    const int bI = gp / 196;
    const int p  = gp - bI * 196;
    const int py = p / 14;
    const int px = p - py * 14;
    const float* abase = img + ((size_t)bI * 3 * 224 + (size_t)py * 16) * 224 + (size_t)px * 16;

    const unsigned alDst = lds_off(&Al[am * LDS_STR + aq * 4]);
    const unsigned blBase = lds_off(&Bl[0]);

    v8f acc[2][6] = {};

    for (int kc = 0; kc < 768; kc += KC) {
        __syncthreads();
        {   // stage A tile (patchify on the fly) via async copy
            int k  = kc + aq * 4;
            int c  = k >> 8;
            int dy = (k >> 4) & 15;
            int dx = k & 15;
            async_load_b128(alDst, abase + ((size_t)c * 224 + dy) * 224 + dx);
        }
#pragma unroll
        for (int it = 0; it < 12; ++it) {   // stage W tile via async copy, coalesced
            int idx = it * 512 + tid;
            int n   = idx >> 3;
            int q   = idx & 7;
            async_load_b128(blBase + (unsigned)(n * LDS_STR + q * 4) * 4u,
                            Wm + (size_t)n * 768 + kc + q * 4);
        }
        wait_async0();                      // own async ops done (LDS writes landed)
        __syncthreads();                    // everyone's tiles visible

#pragma unroll 4
        for (int kk = 0; kk < KC; kk += 4) {
            // A frag: lane -> (M = lanelo, K = kk + laneK + {0,1})
            v2f a0 = *(const v2f*)(&Al[(mrow0 +      lanelo) * LDS_STR + kk + laneK]);
            v2f a1 = *(const v2f*)(&Al[(mrow0 + 16 + lanelo) * LDS_STR + kk + laneK]);
            // B frag: lane -> (N = lanelo, K = kk + laneK + {0,1}); B(k,n) = W[n][k]
            v2f bf[6];
#pragma unroll
            for (int nt = 0; nt < 6; ++nt)
                bf[nt] = *(const v2f*)(&Bl[(ncol0 + nt * 16 + lanelo) * LDS_STR + kk + laneK]);
#pragma unroll
            for (int nt = 0; nt < 6; ++nt) {
                acc[0][nt] = __builtin_amdgcn_wmma_f32_16x16x4_f32(
                    false, a0, false, bf[nt], (short)0, acc[0][nt], false, false);
                acc[1][nt] = __builtin_amdgcn_wmma_f32_16x16x4_f32(
                    false, a1, false, bf[nt], (short)0, acc[1][nt], false, false);
            }
        }
    }

    // ---- epilogue: +bias +positional, per-row partial stats (deterministic) ----
#pragma unroll
    for (int mt = 0; mt < 2; ++mt) {
#pragma unroll
        for (int r = 0; r < 8; ++r) {
            int rowM = mrow0 + mt * 16 + r + laneM8;
            int gpr  = m0g + rowM;
            int bI2  = gpr / 196;
            float tf = (float)(gpr - bI2 * 196 + 1);      // token index t = p+1
            float s = 0.f, q = 0.f;
#pragma unroll
            for (int nt = 0; nt < 6; ++nt) {
                float ang  = tf * fr_r[nt];
                float posv = (lanelo & 1) ? __cosf(ang) : __sinf(ang);
                float x = acc[mt][nt][r] + bi_r[nt] + posv;
                acc[mt][nt][r] = x;
                s += x;  q += x * x;
            }
#pragma unroll
            for (int sh = 1; sh < 16; sh <<= 1) {         // reduce 16 N-lanes (stays in half-wave)
                s += __shfl_xor(s, sh, 32);
                q += __shfl_xor(q, sh, 32);
            }
            if (lanelo == 0) { rowS[rowM * 8 + wn] = s; rowQ[rowM * 8 + wn] = q; }
        }
    }
    __syncthreads();

    if (tid < MT) {    // fixed-order cross-wave sum -> deterministic LN stats
        float S = 0.f, Q = 0.f;
#pragma unroll
        for (int w = 0; w < 8; ++w) { S += rowS[tid * 8 + w]; Q += rowQ[tid * 8 + w]; }
        float mean = S * (1.0f / 768.0f);
        float var  = Q * (1.0f / 768.0f) - mean * mean;
        rowMean[tid] = mean;
        rowInv[tid]  = rsqrtf(var + LN_EPS);
    }
    __syncthreads();

    // ---- normalize + write ----
#pragma unroll
    for (int mt = 0; mt < 2; ++mt) {
#pragma unroll
        for (int r = 0; r < 8; ++r) {
            int rowM = mrow0 + mt * 16 + r + laneM8;
            float mean = rowMean[rowM];
            float inv  = rowInv[rowM];
            int gpr = m0g + rowM;
            int bI2 = gpr / 196;
            int t   = gpr - bI2 * 196 + 1;
            float* orow = out + ((size_t)bI2 * 197 + t) * 768 + ncol0;
#pragma unroll
            for (int nt = 0; nt < 6; ++nt) {
                float y = (acc[mt][nt][r] - mean) * inv * g_r[nt] + be_r[nt];
                orow[nt * 16 + lanelo] = y;
            }
        }
    }
}

// ---------------------------------------------------------------------------
// Class-token rows (t = 0): pos[0][d] = (d odd ? 1 : 0); then LayerNorm.
// One block per image, 256 threads x 3 elements.
// ---------------------------------------------------------------------------
__global__ __launch_bounds__(256)
void vit_cls_ln(const float* __restrict__ cls,
                const float* __restrict__ gamma,
                const float* __restrict__ beta,
                float* __restrict__ out)
{
    __shared__ float redS[8], redQ[8];
    const int b   = blockIdx.x;
    const int tid = threadIdx.x;
    float x[3]; float s = 0.f, q = 0.f;
#pragma unroll
    for (int i = 0; i < 3; ++i) {
        int d = tid + i * 256;
        float v = cls[d] + ((d & 1) ? 1.0f : 0.0f);
        x[i] = v; s += v; q += v * v;
    }
#pragma unroll
    for (int sh = 1; sh < 32; sh <<= 1) { s += __shfl_xor(s, sh, 32); q += __shfl_xor(q, sh, 32); }
    if ((tid & 31) == 0) { redS[tid >> 5] = s; redQ[tid >> 5] = q; }
    __syncthreads();
    float S = 0.f, Q = 0.f;
#pragma unroll
    for (int w = 0; w < 8; ++w) { S += redS[w]; Q += redQ[w]; }
    float mean = S * (1.0f / 768.0f);
    float inv  = rsqrtf(Q * (1.0f / 768.0f) - mean * mean + LN_EPS);
    float* orow = out + (size_t)b * 197 * 768;
#pragma unroll
    for (int i = 0; i < 3; ++i) {
        int d = tid + i * 256;
        orow[d] = (x[i] - mean) * inv * gamma[d] + beta[d];
    }
}

extern "C" void kernel_launch(void* const* d_in, const int* in_sizes, int n_in,
                              void* d_out, int out_size, void* d_ws, size_t ws_size,
                              hipStream_t stream) {
    const float* img   = (const float*)d_in[0];  // [128,3,224,224]
    const float* Wm    = (const float*)d_in[1];  // [768,768]
    const float* bias  = (const float*)d_in[2];  // [768]
    const float* cls   = (const float*)d_in[3];  // [1,768]
    const float* gamma = (const float*)d_in[4];  // [768]
    const float* beta  = (const float*)d_in[5];  // [768]
    float* out = (float*)d_out;                  // [128,197,768]

    (void)in_sizes; (void)n_in; (void)out_size; (void)d_ws; (void)ws_size;

    vit_patch_gemm_ln<<<25088 / MT, 512, 0, stream>>>(img, Wm, bias, gamma, beta, out);
    vit_cls_ln<<<128, 256, 0, stream>>>(cls, gamma, beta, out);
}